// SOMQuantizer_76493367542078
// MI455X (gfx1250) — compile-verified
//
#include <hip/hip_runtime.h>
#include <stdint.h>

#define N_IMG 32
#define C_DIM 512
#define T_DIM 1024
#define M_ROWS (N_IMG * T_DIM)   // 32768
#define K_CODES 1024
#define D_DIM 512
#define W_SOM 32

// ---- output layout (floats), concatenated in reference return order ----
#define SZ_ZQST   ((size_t)N_IMG * C_DIM * T_DIM)      // 16777216
#define OFF_ZQST  ((size_t)0)
#define OFF_COMMIT (SZ_ZQST)
#define OFF_SOM    (SZ_ZQST + 1)
#define OFF_NBR    (SZ_ZQST + 2)
#define SZ_NBR    ((size_t)M_ROWS * 5 * D_DIM)         // 83886080
#define OFF_DIST  (OFF_NBR + SZ_NBR)
#define SZ_DIST   ((size_t)M_ROWS * K_CODES)
#define OFF_K     (OFF_DIST + SZ_DIST)

typedef float v2f __attribute__((ext_vector_type(2)));
typedef float v8f __attribute__((ext_vector_type(8)));
typedef unsigned int u32x4 __attribute__((ext_vector_type(4)));
typedef int i32x4 __attribute__((ext_vector_type(4)));
typedef int i32x8 __attribute__((ext_vector_type(8)));

#if defined(__has_builtin)
#if __has_builtin(__builtin_amdgcn_tensor_load_to_lds)
#define USE_TDM 1
#endif
#endif
#ifndef USE_TDM
#define USE_TDM 0
#endif

#define LDA 132                 // A LDS row stride (floats): 128 + 4 pad
#define LDB 18                  // B LDS row stride (floats): 16 + 2 pad
#define A_TILE_FLOATS (16 * LDA)
#define B_TILE_FLOATS (128 * LDB)
#if USE_TDM
#define NBUF 2
#else
#define NBUF 1
#endif

__device__ __forceinline__ void wait_tensorcnt0() {
#if defined(__has_builtin) && __has_builtin(__builtin_amdgcn_s_wait_tensorcnt)
    __builtin_amdgcn_s_wait_tensorcnt(0);
#else
    asm volatile("s_wait_tensorcnt 0x0" ::: "memory");
#endif
}

#if USE_TDM
// Issue a 2D TDM tile load: tile1 rows of tile0 elements (4B), row stride
// `stride0` elements in memory, with LDS row padding per D# pad fields.
// amdgpu-toolchain (clang-23) 6-arg form: groups 2/3 disabled (zero-filled).
__device__ __forceinline__ void tdm_load_2d(uint32_t lds_off, const float* gptr,
                                            uint32_t tile0, uint32_t tile1,
                                            uint64_t stride0,
                                            uint32_t pad_int, uint32_t pad_amt) {
    uint64_t ga = (uint64_t)(uintptr_t)gptr;
    u32x4 g0;
    g0.x = 1u;                                     // count=1, user descriptor
    g0.y = lds_off;                                // lds_addr [63:32]
    g0.z = (uint32_t)ga;                           // global_addr [95:64]
    g0.w = (uint32_t)(ga >> 32) | (2u << 30);      // global_addr hi | type=2
    const uint32_t td0 = 1u << 20, td1 = 1u << 20; // large dims: no OOB clip
    i32x8 g1;
    g1[0] = (int)((2u << 16) | (1u << 20) | (pad_int << 22) | (pad_amt << 25));
    g1[1] = (int)((td0 & 0xFFFFu) << 16);                    // tensor_dim0 lo
    g1[2] = (int)((td0 >> 16) | ((td1 & 0xFFFFu) << 16));    // dim0 hi|dim1 lo
    g1[3] = (int)((td1 >> 16) | (tile0 << 16));              // dim1 hi|tile0
    g1[4] = (int)tile1;                                      // tile1 | tile2=0
    g1[5] = (int)(uint32_t)stride0;                          // dim0_stride lo
    g1[6] = (int)(uint32_t)(stride0 >> 32);                  // stride hi|s1 lo
    g1[7] = 0;
    i32x4 z4 = {};
    i32x8 z8 = {};
    __builtin_amdgcn_tensor_load_to_lds(g0, g1, z4, z4, z8, 0);
}
#endif

// ---------------- e_sq: one wave per code row -------------------------------
__global__ __launch_bounds__(256) void k_esq(const float* __restrict__ E,
                                             float* __restrict__ esq) {
    int lane = threadIdx.x & 31;
    int row  = blockIdx.x * 8 + (threadIdx.x >> 5);
    const float* r = E + (size_t)row * D_DIM;
    float s = 0.f;
#pragma unroll
    for (int i = 0; i < D_DIM / 32; ++i) { float v = r[lane + i * 32]; s += v * v; }
#pragma unroll
    for (int off = 16; off > 0; off >>= 1) s += __shfl_xor(s, off, 32);
    if (lane == 0) esq[row] = s;
}

// ---------------- z_sq: thread per m, coalesced over t ----------------------
__global__ __launch_bounds__(256) void k_zsq(const float* __restrict__ x,
                                             float* __restrict__ zsq) {
    int m = blockIdx.x * 256 + threadIdx.x;
    int n = m >> 10, t = m & 1023;
    const float* p = x + (size_t)n * C_DIM * T_DIM + t;
    float s = 0.f;
#pragma unroll 4
    for (int c = 0; c < C_DIM; ++c) { float v = p[(size_t)c * T_DIM]; s += v * v; }
    zsq[m] = s;
}

// ---------------- fused distance GEMM: dist = z_sq - 2*(z@E^T) + e_sq -------
// WG tile 128(M) x 128(N), 8 waves as 4(M) x 2(N); wave tile 32x64 (8 frags).
// Tiles staged to LDS by the Tensor Data Mover (double-buffered) when present.
__global__ __launch_bounds__(256) void k_gemm(const float* __restrict__ x,
                                              const float* __restrict__ E,
                                              const float* __restrict__ zsq,
                                              const float* __restrict__ esq,
                                              float* __restrict__ out) {
    __shared__ __align__(16) float As[NBUF * A_TILE_FLOATS]; // As[c][m] stride 132
    __shared__ __align__(16) float Bs[NBUF * B_TILE_FLOATS]; // Bs[n][c] stride 18

    const int tid  = threadIdx.x;
    const int lane = tid & 31;
    const int half = lane >> 4;
    const int l15  = lane & 15;
    const int wave = tid >> 5;
    const int wm   = wave & 3;       // M offset wm*32
    const int wn   = wave >> 2;      // N offset wn*64

    const int n_base = blockIdx.x * 128;
    const int m_base = blockIdx.y * 128;          // 128 | 1024 -> single image
    const int n_img  = m_base >> 10;
    const int t_base = m_base & 1023;
    const float* xb  = x + (size_t)n_img * C_DIM * T_DIM + t_base;
    const float* Eb  = E + (size_t)n_base * D_DIM;

    v8f acc[2][4] = {};
    int buf = 0;

#if USE_TDM
    const uint32_t as_off = (uint32_t)(uintptr_t)As;   // LDS byte offsets
    const uint32_t bs_off = (uint32_t)(uintptr_t)Bs;
    if (wave == 0) {
        // prologue: chunk 0 into buffer 0
        tdm_load_2d(as_off, xb, 128, 16, T_DIM, /*pad 128dw*/6, /*+4dw*/3);
        tdm_load_2d(bs_off, Eb, 16, 128, D_DIM, /*pad 16dw*/3, /*+2dw*/1);
        wait_tensorcnt0();
    }
    __syncthreads();
#endif

    for (int chunk = 0; chunk < D_DIM / 16; ++chunk) {
        const int kc = chunk * 16;
#if USE_TDM
        if (chunk + 1 < D_DIM / 16 && wave == 0) {
            const int kn = kc + 16;
            tdm_load_2d(as_off + (buf ^ 1) * (A_TILE_FLOATS * 4),
                        xb + (size_t)kn * T_DIM, 128, 16, T_DIM, 6, 3);
            tdm_load_2d(bs_off + (buf ^ 1) * (B_TILE_FLOATS * 4),
                        Eb + kn, 16, 128, D_DIM, 3, 1);
        }
#else
        for (int i = tid; i < 16 * 128; i += 256) {           // A stage
            int cl = i >> 7, ml = i & 127;
            As[cl * LDA + ml] = xb[(size_t)(kc + cl) * T_DIM + ml];
        }
        for (int i = tid; i < 16 * 128; i += 256) {           // B stage
            int cl = i & 15, nl = i >> 4;
            Bs[nl * LDB + cl] = Eb[(size_t)nl * D_DIM + kc + cl];
        }
        __syncthreads();
#endif
        const float* Asb = As + buf * A_TILE_FLOATS;
        const float* Bsb = Bs + buf * B_TILE_FLOATS;
#pragma unroll
        for (int ks = 0; ks < 4; ++ks) {
            const int k0 = ks * 4 + 2 * half;   // lanes 0-15: K{0,1}; 16-31: K{2,3}
            v2f a[2], b[4];
#pragma unroll
            for (int mi = 0; mi < 2; ++mi) {
                int ml = wm * 32 + mi * 16 + l15;
                a[mi].x = Asb[k0 * LDA + ml];
                a[mi].y = Asb[(k0 + 1) * LDA + ml];
            }
#pragma unroll
            for (int ni = 0; ni < 4; ++ni) {
                int nl = wn * 64 + ni * 16 + l15;
                b[ni] = *(const v2f*)(Bsb + nl * LDB + k0);   // 8B aligned
            }
#pragma unroll
            for (int mi = 0; mi < 2; ++mi)
#pragma unroll
                for (int ni = 0; ni < 4; ++ni)
                    acc[mi][ni] = __builtin_amdgcn_wmma_f32_16x16x4_f32(
                        false, a[mi], false, b[ni], (short)0, acc[mi][ni],
                        false, false);
        }
#if USE_TDM
        if (wave == 0) wait_tensorcnt0();
        __syncthreads();
        buf ^= 1;
#else
        __syncthreads();
#endif
    }

    // epilogue: dist = zsq[m] - 2*acc + esq[n]; C layout: row = v + 8*half
    float* dist = out + OFF_DIST;
#pragma unroll
    for (int ni = 0; ni < 4; ++ni) {
        int ncol = n_base + wn * 64 + ni * 16 + l15;
        float es = esq[ncol];
#pragma unroll
        for (int mi = 0; mi < 2; ++mi) {
#pragma unroll
            for (int v = 0; v < 8; ++v) {
                int m = m_base + wm * 32 + mi * 16 + v + 8 * half;
                dist[(size_t)m * K_CODES + ncol] = zsq[m] - 2.0f * acc[mi][ni][v] + es;
            }
        }
    }
}

// ---------------- argmin: one wave per row ----------------------------------
__global__ __launch_bounds__(256) void k_argmin(const float* __restrict__ out_dist,
                                                int* __restrict__ kidx,
                                                float* __restrict__ kout) {
    int lane = threadIdx.x & 31;
    int row  = blockIdx.x * 8 + (threadIdx.x >> 5);
    const float* r = out_dist + (size_t)row * K_CODES;
    float best = 3.4e38f; int bi = 0;
#pragma unroll 4
    for (int j = 0; j < K_CODES / 32; ++j) {
        int c = lane + j * 32;
        float v = r[c];
        if (v < best || (v == best && c < bi)) { best = v; bi = c; }
    }
#pragma unroll
    for (int off = 16; off > 0; off >>= 1) {
        float ov = __shfl_xor(best, off, 32);
        int   oi = __shfl_xor(bi,   off, 32);
        if (ov < best || (ov == best && oi < bi)) { best = ov; bi = oi; }
    }
    if (lane == 0) { kidx[row] = bi; kout[row] = (float)bi; }
}

// ---------------- neighbors + loss partials: one block per m ----------------
__global__ __launch_bounds__(256) void k_gather(const float* __restrict__ x,
                                                const float* __restrict__ E,
                                                const int* __restrict__ kidx,
                                                float* __restrict__ out,
                                                float* __restrict__ pcommit,
                                                float* __restrict__ psom) {
    __shared__ float sC[256], sS[256];
    const int m = blockIdx.x, tid = threadIdx.x;
    const int n = m >> 10, t = m & 1023;
    const int k  = kidx[m];
    const int k1 = k / W_SOM, k2 = k % W_SOM;
    int   code[5];
    bool  valid[5];
    code[0] = k;                       valid[0] = true;
    code[1] = (k1 + 1) * W_SOM + k2;   valid[1] = (k1 < W_SOM - 1);   // up
    code[2] = (k1 - 1) * W_SOM + k2;   valid[2] = (k1 > 0);           // down
    code[3] = k1 * W_SOM + (k2 + 1);   valid[3] = (k2 < W_SOM - 1);   // right
    code[4] = k1 * W_SOM + (k2 - 1);   valid[4] = (k2 > 0);           // left

    float cl = 0.f, sl = 0.f;
    float* nbr = out + OFF_NBR + (size_t)m * 5 * D_DIM;
#pragma unroll
    for (int h = 0; h < 2; ++h) {
        int c = tid + h * 256;
        float ze = x[(size_t)n * C_DIM * T_DIM + (size_t)c * T_DIM + t];
#pragma unroll
        for (int j = 0; j < 5; ++j) {
            float q = valid[j] ? E[(size_t)code[j] * D_DIM + c] : 0.f;
            nbr[(size_t)j * D_DIM + c] = q;
            float d = ze - q;
            sl += d * d;
            if (j == 0) cl += d * d;
        }
    }
    sC[tid] = cl; sS[tid] = sl;
    __syncthreads();
    for (int s = 128; s > 0; s >>= 1) {
        if (tid < s) { sC[tid] += sC[tid + s]; sS[tid] += sS[tid + s]; }
        __syncthreads();
    }
    if (tid == 0) { pcommit[m] = sC[0]; psom[m] = sS[0]; }
}

// ---------------- deterministic final loss reduction ------------------------
__global__ __launch_bounds__(256) void k_reduce(const float* __restrict__ pcommit,
                                                const float* __restrict__ psom,
                                                float* __restrict__ out) {
    __shared__ float sC[256], sS[256];
    int tid = threadIdx.x;
    float c = 0.f, s = 0.f;
    for (int i = tid; i < M_ROWS; i += 256) { c += pcommit[i]; s += psom[i]; }
    sC[tid] = c; sS[tid] = s;
    __syncthreads();
    for (int st = 128; st > 0; st >>= 1) {
        if (tid < st) { sC[tid] += sC[tid + st]; sS[tid] += sS[tid + st]; }
        __syncthreads();
    }
    if (tid == 0) {
        out[OFF_COMMIT] = 2.0f * sC[0] / (float)((size_t)M_ROWS * D_DIM);
        out[OFF_SOM]    = sS[0] / (float)((size_t)M_ROWS * 5 * D_DIM);
    }
}

// ---------------- z_q_st: gather E rows back into (N,C,T) layout ------------
__global__ __launch_bounds__(256) void k_zqst(const float* __restrict__ E,
                                              const int* __restrict__ kidx,
                                              float* __restrict__ out) {
    size_t e = (size_t)blockIdx.x * 256 + threadIdx.x;
    int t = (int)(e & 1023);
    int c = (int)((e >> 10) & 511);
    int n = (int)(e >> 19);
    int m = n * T_DIM + t;
    out[OFF_ZQST + e] = E[(size_t)kidx[m] * D_DIM + c];
}

extern "C" void kernel_launch(void* const* d_in, const int* in_sizes, int n_in,
                              void* d_out, int out_size, void* d_ws, size_t ws_size,
                              hipStream_t stream) {
    const float* x = (const float*)d_in[0];         // (32, 512, 1024)
    const float* E = (const float*)d_in[1];         // (32, 32, 512) -> (1024, 512)
    float* out = (float*)d_out;

    char* w = (char*)d_ws;
    float* esq  = (float*)(w + 0);          // 1024 f
    float* zsq  = (float*)(w + 4096);       // 32768 f
    int*   kidx = (int*)  (w + 135168);     // 32768 i32
    float* pcm  = (float*)(w + 266240);     // 32768 f
    float* psm  = (float*)(w + 397312);     // 32768 f

    k_esq   <<<K_CODES / 8, 256, 0, stream>>>(E, esq);
    k_zsq   <<<M_ROWS / 256, 256, 0, stream>>>(x, zsq);
    k_gemm  <<<dim3(K_CODES / 128, M_ROWS / 128), 256, 0, stream>>>(x, E, zsq, esq, out);
    k_argmin<<<M_ROWS / 8, 256, 0, stream>>>(out + OFF_DIST, kidx, out + OFF_K);
    k_gather<<<M_ROWS, 256, 0, stream>>>(x, E, kidx, out, pcm, psm);
    k_reduce<<<1, 256, 0, stream>>>(pcm, psm, out);
    k_zqst  <<<(unsigned)(SZ_ZQST / 256), 256, 0, stream>>>(E, kidx, out);
}